// FGL_useless_27376121544987
// MI455X (gfx1250) — compile-verified
//
#include <hip/hip_runtime.h>

#define INC   32
#define INN   50000
#define OUTC  32
#define OUTN  8192
#define MAXD  16
#define NB    8           // batch N
#define RCOLS (NB * INC)  // 256 embedding columns

typedef float v2f __attribute__((ext_vector_type(2)));
typedef float v8f __attribute__((ext_vector_type(8)));

// ---------------------------------------------------------------------------
// Kernel 1: transpose x [256][50000] -> ew [50000][256] (tiled via LDS).
// Coalesced reads along INN, coalesced writes along the 256 row-columns.
// ---------------------------------------------------------------------------
__global__ void __launch_bounds__(256) xpose_kernel(const float* __restrict__ x,
                                                    float* __restrict__ ew) {
    __shared__ float tile[32][33];
    const int i0 = blockIdx.x * 32;   // INN tile
    const int r0 = blockIdx.y * 32;   // row (n*INC+c) tile
    const int tx = threadIdx.x;       // 0..31
    const int ty = threadIdx.y;       // 0..7

#pragma unroll
    for (int k = 0; k < 4; ++k) {
        const int r = r0 + ty + k * 8;      // always < 256
        const int i = i0 + tx;
        tile[ty + k * 8][tx] = (i < INN) ? x[(size_t)r * INN + i] : 0.0f;
    }
    __syncthreads();
#pragma unroll
    for (int k = 0; k < 4; ++k) {
        const int i = i0 + ty + k * 8;
        if (i < INN)
            ew[(size_t)i * RCOLS + (r0 + tx)] = tile[tx][ty + k * 8];
    }
}

// ---------------------------------------------------------------------------
// Kernel 2: fused gather + masked mean-pool + f32 WMMA GEMM + bias.
// One block = 16 output nodes, 256 threads = 8 waves (one wave per batch n).
// ---------------------------------------------------------------------------
#define PPITCH 260   // 16 rows; stride%64==4 -> A-fragment column reads conflict-free
#define OPITCH 17    // output staging pad

__global__ void __launch_bounds__(256) fused_gcn_kernel(
    const float* __restrict__ ew, const int* __restrict__ Aidx,
    const float* __restrict__ mask, const float* __restrict__ weight,
    const float* __restrict__ bias, float* __restrict__ out) {

    __shared__ float P[16 * PPITCH];           // pooled tile [o_local][r=n*32+c]
    __shared__ float O[NB * OUTC * OPITCH];    // output staging [n][dOut][o_local]

    const int tid    = threadIdx.x;
    const int wave   = tid >> 5;
    const int lane   = tid & 31;
    const int o_base = blockIdx.x * 16;

    // ---- Phase A: gather 16 neighbor rows (1KB each, contiguous) per node ----
#pragma unroll
    for (int j = 0; j < 2; ++j) {
        const int o_local = wave * 2 + j;
        const int o       = o_base + o_local;
        float a0 = 0.f, a1 = 0.f, a2 = 0.f, a3 = 0.f;
        float b0 = 0.f, b1 = 0.f, b2 = 0.f, b3 = 0.f;
        for (int d = 0; d < MAXD; ++d) {
            const int   idx = Aidx[o * MAXD + d];
            const float m   = mask[o * MAXD + d];
            const float4* row = (const float4*)(ew + (size_t)idx * RCOLS);
            const float4 r0 = row[lane];        // r = lane*4 .. lane*4+3
            const float4 r1 = row[lane + 32];   // r = 128 + lane*4 ..
            a0 += m * r0.x; a1 += m * r0.y; a2 += m * r0.z; a3 += m * r0.w;
            b0 += m * r1.x; b1 += m * r1.y; b2 += m * r1.z; b3 += m * r1.w;
        }
        float* p = P + o_local * PPITCH;
        p[lane * 4 + 0]       = a0; p[lane * 4 + 1]       = a1;
        p[lane * 4 + 2]       = a2; p[lane * 4 + 3]       = a3;
        p[128 + lane * 4 + 0] = b0; p[128 + lane * 4 + 1] = b1;
        p[128 + lane * 4 + 2] = b2; p[128 + lane * 4 + 3] = b3;
    }
    __syncthreads();

    // ---- Phase B: D[16 o][32 dOut] = P[:, n*32 + c] * weight[c][dOut] ----
    // V_WMMA_F32_16X16X4_F32, K=32 chained as 8 steps; wave 'n' owns batch n.
    const int n    = wave;
    const int mrow = lane & 15;               // M row (A) / N col (B)
    const int ksel = (lane < 16) ? 0 : 2;     // 16x4 f32 A layout: K0/K1 vs K2/K3
    const int moff = (lane < 16) ? 0 : 8;

#pragma unroll
    for (int h = 0; h < 2; ++h) {             // dOut halves 0..15 / 16..31
        v8f acc = {0.f, 0.f, 0.f, 0.f, 0.f, 0.f, 0.f, 0.f};
#pragma unroll
        for (int kb = 0; kb < 8; ++kb) {
            const int k0 = kb * 4 + ksel;
            v2f av, bv;
            av.x = P[mrow * PPITCH + n * 32 + k0];
            av.y = P[mrow * PPITCH + n * 32 + k0 + 1];
            bv.x = weight[(k0    ) * OUTC + h * 16 + mrow];
            bv.y = weight[(k0 + 1) * OUTC + h * 16 + mrow];
            acc = __builtin_amdgcn_wmma_f32_16x16x4_f32(
                /*neg_a=*/false, av, /*neg_b=*/false, bv,
                /*c_mod=*/(short)0, acc, /*reuse_a=*/false, /*reuse_b=*/false);
        }
        // C/D layout: VGPR j -> M = j (lanes 0-15) / j+8 (lanes 16-31), N = lane&15
#pragma unroll
        for (int jj = 0; jj < 8; ++jj)
            O[(n * OUTC + h * 16 + mrow) * OPITCH + (jj + moff)] = acc[jj];
    }
    __syncthreads();

    // ---- Phase C: bias add + store; thread t owns one (n,dOut) row of 16 o ----
    {
        const int nn = tid >> 5;   // 0..7
        const int d  = tid & 31;   // 0..31
        const float* orow = O + (nn * OUTC + d) * OPITCH;
        const float* brow = bias + (size_t)d * OUTN + o_base;
        float* dst = out + (size_t)(nn * OUTC + d) * OUTN + o_base;
#pragma unroll
        for (int j2 = 0; j2 < 16; ++j2)
            dst[j2] = orow[j2] + brow[j2];
    }
}

// ---------------------------------------------------------------------------
extern "C" void kernel_launch(void* const* d_in, const int* in_sizes, int n_in,
                              void* d_out, int out_size, void* d_ws, size_t ws_size,
                              hipStream_t stream) {
    (void)in_sizes; (void)n_in; (void)out_size; (void)ws_size;
    const float* x      = (const float*)d_in[0];   // [8][32][50000]
    const int*   Aidx   = (const int*)  d_in[1];   // [8192][16]
    const float* mask   = (const float*)d_in[2];   // [8192][16][1]
    const float* weight = (const float*)d_in[3];   // [32][32]
    const float* bias   = (const float*)d_in[4];   // [32][8192]
    float*       out    = (float*)d_out;           // [8][32][8192]
    float*       ew     = (float*)d_ws;            // [50000][256] = 51.2 MB scratch

    dim3 tb(32, 8);
    dim3 tg((INN + 31) / 32, RCOLS / 32);
    xpose_kernel<<<tg, tb, 0, stream>>>(x, ew);

    fused_gcn_kernel<<<OUTN / 16, 256, 0, stream>>>(ew, Aidx, mask, weight, bias, out);
}